// self_attention_25795573580011
// MI455X (gfx1250) — compile-verified
//
#include <hip/hip_runtime.h>

typedef __attribute__((ext_vector_type(2))) float v2f;
typedef __attribute__((ext_vector_type(8))) float v8f;

#define CH   64
#define CF   8
#define NTOK 4096   // 64*64 tokens per batch
#define NB   8

// ---------------------------------------------------------------------------
// Async global->LDS copy support (CDNA5 GLOBAL_LOAD_ASYNC_TO_LDS_*, ASYNCcnt)
// Builtin signature (from hipcc diagnostics):
//   b32 : (int  addrspace(1)* src, int  addrspace(3)* dst, imm off, imm cpol)
//   b128: (int4 addrspace(1)* src, int4 addrspace(3)* dst, imm off, imm cpol)
//   where int4 is a gcc-vector: int __attribute__((vector_size(16)))
// ---------------------------------------------------------------------------
#if defined(__AMDGCN__) && defined(__gfx1250__) && \
    __has_builtin(__builtin_amdgcn_global_load_async_to_lds_b128) && \
    __has_builtin(__builtin_amdgcn_global_load_async_to_lds_b32)
#define USE_ASYNC 1
#else
#define USE_ASYNC 0
#endif

#if USE_ASYNC
typedef int v4i_gcc __attribute__((vector_size(16)));
typedef __attribute__((address_space(1))) int     g_i32;
typedef __attribute__((address_space(3))) int     l_i32;
typedef __attribute__((address_space(1))) v4i_gcc g_v4i;
typedef __attribute__((address_space(3))) v4i_gcc l_v4i;

__device__ __forceinline__ void async_copy_b32(const float* g, float* l) {
  __builtin_amdgcn_global_load_async_to_lds_b32((g_i32*)g, (l_i32*)l, 0, 0);
}
__device__ __forceinline__ void async_copy_b128(const float* g, float* l) {
  __builtin_amdgcn_global_load_async_to_lds_b128((g_v4i*)g, (l_v4i*)l, 0, 0);
}
#if __has_builtin(__builtin_amdgcn_s_wait_asynccnt)
#define WAIT_ASYNC(n) __builtin_amdgcn_s_wait_asynccnt(n)
#else
#define WAIT_ASYNC(n) asm volatile("s_wait_asynccnt " #n ::: "memory")
#endif
#else
#define WAIT_ASYNC(n)
#endif

// ---------------------------------------------------------------------------
// Kernel 1: projections.  FG = X @ [Kf|Kg]  ([rows,16]),  H = X @ Kh ([rows,64])
// One wave per 16-row tile, 4 waves per block. Full-fp32 WMMA (16x16x4).
// ---------------------------------------------------------------------------
__global__ __launch_bounds__(128) void proj_kernel(
    const float* __restrict__ x,
    const float* __restrict__ kf,
    const float* __restrict__ kg,
    const float* __restrict__ kh,
    float* __restrict__ FG,
    float* __restrict__ H)
{
  __shared__ float sKfg[CH][16];
  __shared__ float sKh[CH][CH];
  const int tid = threadIdx.x;

  for (int i = tid; i < CH * 16; i += 128) {
    int c = i >> 4, j = i & 15;
    sKfg[c][j] = (j < 8) ? kf[c * 8 + j] : kg[c * 8 + (j - 8)];
  }
  for (int i = tid; i < CH * CH; i += 128) {
    sKh[i >> 6][i & 63] = kh[i];
  }
  __syncthreads();

  const int wave = tid >> 5;
  const int lane = tid & 31;
  const int half = lane >> 4;      // which 16-lane half
  const int n    = lane & 15;      // column index / A-row index
  const int kb   = half * 2;       // K offset within a 4-chunk

  const int tilebase = blockIdx.x * 64 + wave * 16;
  const int arow     = tilebase + n;

  // A operands: 16 chunks of K=4 covering the 64 input channels
  v2f a[16];
  const float* xr = x + (size_t)arow * CH;
  #pragma unroll
  for (int kc = 0; kc < 16; ++kc) {
    a[kc].x = xr[kc * 4 + kb + 0];
    a[kc].y = xr[kc * 4 + kb + 1];
  }

  v8f cFG = {};
  #pragma unroll
  for (int kc = 0; kc < 16; ++kc) {
    v2f b;
    b.x = sKfg[kc * 4 + kb + 0][n];
    b.y = sKfg[kc * 4 + kb + 1][n];
    cFG = __builtin_amdgcn_wmma_f32_16x16x4_f32(false, a[kc], false, b,
                                                (short)0, cFG, false, false);
  }

  v8f cH[4] = {{}, {}, {}, {}};
  #pragma unroll
  for (int nt = 0; nt < 4; ++nt) {
    #pragma unroll
    for (int kc = 0; kc < 16; ++kc) {
      v2f b;
      b.x = sKh[kc * 4 + kb + 0][nt * 16 + n];
      b.y = sKh[kc * 4 + kb + 1][nt * 16 + n];
      cH[nt] = __builtin_amdgcn_wmma_f32_16x16x4_f32(false, a[kc], false, b,
                                                     (short)0, cH[nt], false, false);
    }
  }

  // C/D layout: row = tilebase + v + 8*half, col = n
  const int obase = tilebase + half * 8;
  #pragma unroll
  for (int v = 0; v < 8; ++v) {
    const int orow = obase + v;
    FG[(size_t)orow * 16 + n] = cFG[v];
    #pragma unroll
    for (int nt = 0; nt < 4; ++nt)
      H[(size_t)orow * CH + nt * 16 + n] = cH[nt][v];
  }
}

// ---------------------------------------------------------------------------
// Kernel 2: fused flash attention.  Each wave owns 16 query rows; workgroup
// (4 waves = 64 query rows) stages each 16-row key tile's F (16x8) and
// H (16x64) in double-buffered LDS via async global->LDS copies, prefetching
// tile j+1 while computing tile j.  Online softmax; O accumulated via WMMA.
// ---------------------------------------------------------------------------
__global__ __launch_bounds__(128) void attn_kernel(
    const float* __restrict__ x,
    const float* __restrict__ FG,
    const float* __restrict__ H,
    const float* __restrict__ gamma_p,
    float* __restrict__ out)
{
  __shared__ float sF[2][16][CF];
  __shared__ float sH[2][16][CH];
  __shared__ float sP[4][16][16];   // per-wave P tile (C-layout -> A-layout relay)

  const int tid  = threadIdx.x;
  const int wave = tid >> 5;
  const int lane = tid & 31;
  const int half = lane >> 4;
  const int n    = lane & 15;
  const int kb   = half * 2;

  const int bbase = blockIdx.y * NTOK;
  const int qtile = blockIdx.x * 64 + wave * 16;

  // per-thread staging slots: 3 copies per tile (1x b32 F, 2x b128 H)
  const int skr = tid >> 3;          // key row handled by this thread
  const int sc  = tid & 7;           // sub-column

  // G as WMMA-A operands (channels 0..7 of g live in FG cols 8..15)
  const float* gr = FG + (size_t)(bbase + qtile + n) * 16;
  v2f gA0, gA1;
  gA0.x = gr[8  + kb]; gA0.y = gr[8  + kb + 1];
  gA1.x = gr[12 + kb]; gA1.y = gr[12 + kb + 1];

  float row_m[8], row_l[8];
  v8f O[4] = {{}, {}, {}, {}};
  #pragma unroll
  for (int v = 0; v < 8; ++v) { row_m[v] = -1e30f; row_l[v] = 0.0f; }

  // ---- stage tile 0 into buffer 0 ----
  {
    const size_t rb = (size_t)(bbase + skr);
    const float* fp = FG + rb * 16 + sc;
    const float* hp = H + rb * CH + sc * 8;
#if USE_ASYNC
    async_copy_b32(fp, &sF[0][skr][sc]);
    async_copy_b128(hp + 0, &sH[0][skr][sc * 8 + 0]);
    async_copy_b128(hp + 4, &sH[0][skr][sc * 8 + 4]);
#else
    sF[0][skr][sc] = *fp;
    *(float4*)&sH[0][skr][sc * 8 + 0] = *(const float4*)(hp + 0);
    *(float4*)&sH[0][skr][sc * 8 + 4] = *(const float4*)(hp + 4);
#endif
  }

  const int NT = NTOK / 16;
  for (int j = 0; j < NT; ++j) {
    const int buf = j & 1;
    // all waves are done reading buffer (j+1)&1 (used in iteration j-1)
    __syncthreads();
    if (j + 1 < NT) {
      const int nbuf = buf ^ 1;
      const size_t rb = (size_t)(bbase + (j + 1) * 16 + skr);
      const float* fp = FG + rb * 16 + sc;
      const float* hp = H + rb * CH + sc * 8;
#if USE_ASYNC
      async_copy_b32(fp, &sF[nbuf][skr][sc]);
      async_copy_b128(hp + 0, &sH[nbuf][skr][sc * 8 + 0]);
      async_copy_b128(hp + 4, &sH[nbuf][skr][sc * 8 + 4]);
#else
      sF[nbuf][skr][sc] = *fp;
      *(float4*)&sH[nbuf][skr][sc * 8 + 0] = *(const float4*)(hp + 0);
      *(float4*)&sH[nbuf][skr][sc * 8 + 4] = *(const float4*)(hp + 4);
#endif
      WAIT_ASYNC(3);   // tile j's 3 copies (issued last iter) are complete
    } else {
      WAIT_ASYNC(0);   // drain: last tile's copies complete
    }
    // every wave has waited for its share of tile j -> tile j visible to all
    __syncthreads();

    // S tile = G (16x8) @ F^T (8x16): two K=4 WMMAs
    v2f fB0, fB1;
    fB0.x = sF[buf][n][kb + 0]; fB0.y = sF[buf][n][kb + 1];
    fB1.x = sF[buf][n][4 + kb]; fB1.y = sF[buf][n][4 + kb + 1];
    v8f s = {};
    s = __builtin_amdgcn_wmma_f32_16x16x4_f32(false, gA0, false, fB0, (short)0, s, false, false);
    s = __builtin_amdgcn_wmma_f32_16x16x4_f32(false, gA1, false, fB1, (short)0, s, false, false);

    // online softmax: row = v + 8*half, its 16 columns live across one half-wave
    #pragma unroll
    for (int v = 0; v < 8; ++v) {
      float sv  = s[v];
      float red = sv;
      red = fmaxf(red, __shfl_xor(red, 1, 32));
      red = fmaxf(red, __shfl_xor(red, 2, 32));
      red = fmaxf(red, __shfl_xor(red, 4, 32));
      red = fmaxf(red, __shfl_xor(red, 8, 32));
      const float mnew  = fmaxf(row_m[v], red);
      const float alpha = __expf(row_m[v] - mnew);
      const float p     = __expf(sv - mnew);
      float rs = p;
      rs += __shfl_xor(rs, 1, 32);
      rs += __shfl_xor(rs, 2, 32);
      rs += __shfl_xor(rs, 4, 32);
      rs += __shfl_xor(rs, 8, 32);
      row_l[v] = row_l[v] * alpha + rs;
      row_m[v] = mnew;
      sP[wave][half * 8 + v][n] = p;          // C-layout scatter
      O[0][v] *= alpha; O[1][v] *= alpha; O[2][v] *= alpha; O[3][v] *= alpha;
    }

    // O += P (16x16) @ Htile (16x64): 4 k-chunks x 4 column tiles
    #pragma unroll
    for (int kc = 0; kc < 4; ++kc) {
      v2f pA;
      pA.x = sP[wave][n][kc * 4 + kb + 0];    // A-layout gather
      pA.y = sP[wave][n][kc * 4 + kb + 1];
      #pragma unroll
      for (int nt = 0; nt < 4; ++nt) {
        v2f hB;
        hB.x = sH[buf][kc * 4 + kb + 0][nt * 16 + n];
        hB.y = sH[buf][kc * 4 + kb + 1][nt * 16 + n];
        O[nt] = __builtin_amdgcn_wmma_f32_16x16x4_f32(false, pA, false, hB,
                                                      (short)0, O[nt], false, false);
      }
    }
  }

  const float gamma = gamma_p[0];
  #pragma unroll
  for (int v = 0; v < 8; ++v) {
    const int   grow  = bbase + qtile + half * 8 + v;
    const float inv_l = 1.0f / row_l[v];
    #pragma unroll
    for (int nt = 0; nt < 4; ++nt) {
      const int ch = nt * 16 + n;
      out[(size_t)grow * CH + ch] =
          gamma * (O[nt][v] * inv_l) + x[(size_t)grow * CH + ch];
    }
  }
}

// ---------------------------------------------------------------------------
extern "C" void kernel_launch(void* const* d_in, const int* in_sizes, int n_in,
                              void* d_out, int out_size, void* d_ws, size_t ws_size,
                              hipStream_t stream) {
  (void)in_sizes; (void)n_in; (void)out_size; (void)ws_size;
  const float* x     = (const float*)d_in[0];
  const float* kf    = (const float*)d_in[1];
  const float* kg    = (const float*)d_in[2];
  const float* kh    = (const float*)d_in[3];
  const float* gamma = (const float*)d_in[4];
  float*       out   = (float*)d_out;

  float* FG = (float*)d_ws;                       // [NB*NTOK, 16]  = 2 MB
  float* Hb = FG + (size_t)NB * NTOK * 16;        // [NB*NTOK, 64]  = 8 MB

  proj_kernel<<<dim3(NB * NTOK / 64), 128, 0, stream>>>(x, kf, kg, kh, FG, Hb);
  attn_kernel<<<dim3(NTOK / 64, NB), 128, 0, stream>>>(x, FG, Hb, gamma, out);
}